// RGCN_59107339927815
// MI455X (gfx1250) — compile-verified
//
#include <hip/hip_runtime.h>
#include <hip/hip_bf16.h>

typedef __attribute__((ext_vector_type(2))) float v2f;
typedef __attribute__((ext_vector_type(8))) float v8f;

#define NU 100000
#define NI 50000
#define EDG 500000
#define DIM 128
#define MTILES 5          // 5 x 16 = 80 rows per block; 80 divides NU and NI
#define NEG_SLOPE 0.01f

// ---------------------------------------------------------------------------
// GEMM: Y[n,128] = X[n,128] @ W[128,128], f32 WMMA 16x16x4.
// Block = 256 threads = 8 waves. Block computes an 80(M) x 128(N) strip:
// wave w owns N-tile [w*16, w*16+16), loops over 5 M-tiles.
// W staged once per block into LDS (64 KB). n must be a multiple of 80.
// ---------------------------------------------------------------------------
__global__ void __launch_bounds__(256)
gemm128_wmma(const float* __restrict__ X, const float* __restrict__ W,
             float* __restrict__ Y, int n) {
  __shared__ float sW[DIM * DIM];  // 64 KB

  // cooperative stage of W into LDS as float4
  {
    const float4* Wv = (const float4*)W;
    float4* sWv = (float4*)sW;
    for (int i = threadIdx.x; i < (DIM * DIM) / 4; i += 256) sWv[i] = Wv[i];
  }
  __syncthreads();

  const int lane = threadIdx.x & 31;
  const int wave = threadIdx.x >> 5;
  const int row  = lane & 15;          // M for A, N for B/C/D
  const int kh   = (lane >> 4) << 1;   // K sub-offset: 0 (lanes 0-15) / 2 (lanes 16-31)
  const int col  = (wave << 4) + row;  // N column in [0,128)
  const int mHalf = (lane >> 4) << 3;  // D: M offset 0 / 8

  for (int mt = 0; mt < MTILES; ++mt) {
    const int mBase = blockIdx.x * (16 * MTILES) + mt * 16;
    const float* xr = X + (size_t)(mBase + row) * DIM;

    v8f acc = {};
    #pragma unroll 4
    for (int k0 = 0; k0 < DIM; k0 += 4) {
      // A 16x4 f32: lane holds X[row][k0+kh], X[row][k0+kh+1]
      v2f a;
      a.x = xr[k0 + kh];
      a.y = xr[k0 + kh + 1];
      // B 4x16 f32: lane holds W[k0+kh][col], W[k0+kh+1][col]
      v2f b;
      b.x = sW[(k0 + kh) * DIM + col];
      b.y = sW[(k0 + kh + 1) * DIM + col];
      acc = __builtin_amdgcn_wmma_f32_16x16x4_f32(
          /*neg_a=*/false, a, /*neg_b=*/false, b,
          /*c_mod=*/(short)0, acc, /*reuse_a=*/false, /*reuse_b=*/false);
    }

    // D 16x16 f32: VGPR r -> M = r + (lane<16 ? 0 : 8), N = col. Unconditional.
    float* yr = Y + (size_t)(mBase + mHalf) * DIM + col;
    #pragma unroll
    for (int r = 0; r < 8; ++r) {
      yr[(size_t)r * DIM] = acc[r];
    }
  }
}

// ---------------------------------------------------------------------------
// zero fill (float4 granularity)
// ---------------------------------------------------------------------------
__global__ void zero4_kernel(float4* __restrict__ p, int n4) {
  int t = blockIdx.x * blockDim.x + threadIdx.x;
  if (t < n4) p[t] = make_float4(0.f, 0.f, 0.f, 0.f);
}

// ---------------------------------------------------------------------------
// degree count: deg[dst[e]] += 1
// ---------------------------------------------------------------------------
__global__ void deg_kernel(const int* __restrict__ dst, float* __restrict__ deg, int e) {
  int t = blockIdx.x * blockDim.x + threadIdx.x;
  if (t < e) atomicAdd(&deg[dst[t]], 1.0f);
}

// ---------------------------------------------------------------------------
// scatter-add: one wave per edge. Lane loads float4 of h[src], atomically
// adds into agg[dst]. Atomics resolve in L2 (outputs are L2-resident).
// ---------------------------------------------------------------------------
__global__ void __launch_bounds__(256)
scatter_kernel(const float* __restrict__ h, const int* __restrict__ src,
               const int* __restrict__ dst, float* __restrict__ agg, int e) {
  int t = blockIdx.x * blockDim.x + threadIdx.x;
  int edge = t >> 5;
  if (edge >= e) return;
  int lane = t & 31;
  int s = src[edge];
  int d = dst[edge];
  const float4 v = ((const float4*)(h + (size_t)s * DIM))[lane];
  float* o = agg + (size_t)d * DIM + lane * 4;
  atomicAdd(o + 0, v.x);
  atomicAdd(o + 1, v.y);
  atomicAdd(o + 2, v.z);
  atomicAdd(o + 3, v.w);
}

// ---------------------------------------------------------------------------
// finalize: out = agg / max(deg,1) + bias, optional leaky-relu. In place.
// ---------------------------------------------------------------------------
__global__ void finalize_kernel(float* __restrict__ agg, const float* __restrict__ deg,
                                const float* __restrict__ bias, int n, int relu) {
  int t = blockIdx.x * blockDim.x + threadIdx.x;
  if (t >= n * DIM) return;
  int i = t >> 7;
  int c = t & (DIM - 1);
  float dg = deg[i];
  dg = dg < 1.0f ? 1.0f : dg;
  float v = agg[t] / dg + bias[c];
  if (relu) v = v >= 0.0f ? v : NEG_SLOPE * v;
  agg[t] = v;
}

// ---------------------------------------------------------------------------
extern "C" void kernel_launch(void* const* d_in, const int* in_sizes, int n_in,
                              void* d_out, int out_size, void* d_ws, size_t ws_size,
                              hipStream_t stream) {
  const float* x_user   = (const float*)d_in[0];
  const float* x_item   = (const float*)d_in[1];
  const int*   src_rates= (const int*)d_in[2];
  const int*   dst_rates= (const int*)d_in[3];
  const int*   src_rev  = (const int*)d_in[4];
  const int*   dst_rev  = (const int*)d_in[5];
  const float* W1_rates = (const float*)d_in[6];
  const float* b1_rates = (const float*)d_in[7];
  const float* W1_rev   = (const float*)d_in[8];
  const float* b1_rev   = (const float*)d_in[9];
  const float* W2_rates = (const float*)d_in[10];
  const float* b2_rates = (const float*)d_in[11];
  const float* W2_rev   = (const float*)d_in[12];
  const float* b2_rev   = (const float*)d_in[13];

  float* ws       = (float*)d_ws;
  float* t_user   = ws;                              // NU*128
  float* t_item   = t_user + (size_t)NU * DIM;       // NI*128
  float* agg_user = t_item + (size_t)NI * DIM;       // NU*128
  float* agg_item = agg_user + (size_t)NU * DIM;     // NI*128
  float* deg_user = agg_item + (size_t)NI * DIM;     // NU
  float* deg_item = deg_user + NU;                   // NI

  float* o_user = (float*)d_out;                     // NU*128
  float* o_item = o_user + (size_t)NU * DIM;         // NI*128

  const int scatterBlocks = (EDG * 32) / 256;        // 62500
  const int degBlocks = (EDG + 255) / 256;
  const int ROWS = 16 * MTILES;                      // 80

  // ---- layer 1 GEMMs
  gemm128_wmma<<<NU / ROWS, 256, 0, stream>>>(x_user, W1_rates, t_user, NU);
  gemm128_wmma<<<NI / ROWS, 256, 0, stream>>>(x_item, W1_rev, t_item, NI);

  // ---- zero agg + deg (contiguous region in ws)
  {
    size_t zf = (size_t)(NU + NI) * DIM + NU + NI;   // floats, multiple of 4
    int n4 = (int)(zf / 4);
    zero4_kernel<<<(n4 + 255) / 256, 256, 0, stream>>>((float4*)agg_user, n4);
  }

  // ---- degrees (shared by both layers)
  deg_kernel<<<degBlocks, 256, 0, stream>>>(dst_rates, deg_item, EDG);
  deg_kernel<<<degBlocks, 256, 0, stream>>>(dst_rev, deg_user, EDG);

  // ---- layer 1 scatter + finalize (leaky relu)
  scatter_kernel<<<scatterBlocks, 256, 0, stream>>>(t_user, src_rates, dst_rates, agg_item, EDG);
  scatter_kernel<<<scatterBlocks, 256, 0, stream>>>(t_item, src_rev, dst_rev, agg_user, EDG);
  finalize_kernel<<<(NI * DIM + 255) / 256, 256, 0, stream>>>(agg_item, deg_item, b1_rates, NI, 1);
  finalize_kernel<<<(NU * DIM + 255) / 256, 256, 0, stream>>>(agg_user, deg_user, b1_rev, NU, 1);

  // ---- layer 2 GEMMs (reuse t buffers)
  gemm128_wmma<<<NU / ROWS, 256, 0, stream>>>(agg_user, W2_rates, t_user, NU);
  gemm128_wmma<<<NI / ROWS, 256, 0, stream>>>(agg_item, W2_rev, t_item, NI);

  // ---- zero output, scatter into d_out
  {
    int n4 = (int)((size_t)(NU + NI) * DIM / 4);
    zero4_kernel<<<(n4 + 255) / 256, 256, 0, stream>>>((float4*)d_out, n4);
  }
  scatter_kernel<<<scatterBlocks, 256, 0, stream>>>(t_user, src_rates, dst_rates, o_item, EDG);
  scatter_kernel<<<scatterBlocks, 256, 0, stream>>>(t_item, src_rev, dst_rev, o_user, EDG);

  // ---- finalize outputs (no relu on layer 2)
  finalize_kernel<<<(NU * DIM + 255) / 256, 256, 0, stream>>>(o_user, deg_user, b2_rev, NU, 0);
  finalize_kernel<<<(NI * DIM + 255) / 256, 256, 0, stream>>>(o_item, deg_item, b2_rates, NI, 0);
}